// GCN_5841155522892
// MI455X (gfx1250) — compile-verified
//
#include <hip/hip_runtime.h>

typedef __attribute__((ext_vector_type(2))) float v2f;
typedef __attribute__((ext_vector_type(8))) float v8f;

// ---------------------------------------------------------------------------
// Phase 1: pack feature[N,3] -> fpad[N] = {f0,f1,f2,1.0f}
// (w = 1.0 lets the edge kernel accumulate degree with the same atomic path)
// ---------------------------------------------------------------------------
__global__ __launch_bounds__(256) void gcn_pack_kernel(
    const float* __restrict__ feature, float4* __restrict__ fpad, int N) {
  int i = blockIdx.x * blockDim.x + threadIdx.x;
  if (i >= N) return;
  const float* f = feature + (size_t)i * 3;
  fpad[i] = make_float4(f[0], f[1], f[2], 1.0f);
}

// ---------------------------------------------------------------------------
// Phase 2: edge scatter. One thread per edge.
//   s1pad[d] += {feature[s], 1.0}   (four contiguous f32 atomics into L2)
// src/dst are streamed once -> non-temporal loads so they don't evict the
// hot 16MB gather set / 16MB scatter set from the 192MB L2.
// ---------------------------------------------------------------------------
__global__ __launch_bounds__(256) void gcn_edge_kernel(
    const float4* __restrict__ fpad, const int* __restrict__ src,
    const int* __restrict__ dst, float* __restrict__ s1, int E) {
  int e = blockIdx.x * blockDim.x + threadIdx.x;
  if (e >= E) return;
  int s = __builtin_nontemporal_load(src + e);
  int d = __builtin_nontemporal_load(dst + e);
  float4 f = fpad[s];
  float* p = s1 + (size_t)d * 4;
  unsafeAtomicAdd(p + 0, f.x);
  unsafeAtomicAdd(p + 1, f.y);
  unsafeAtomicAdd(p + 2, f.z);
  unsafeAtomicAdd(p + 3, f.w);  // degree
}

// Fallback edge kernel if workspace is too small for the packed feature copy.
__global__ __launch_bounds__(256) void gcn_edge_kernel_direct(
    const float* __restrict__ feature, const int* __restrict__ src,
    const int* __restrict__ dst, float* __restrict__ s1, int E) {
  int e = blockIdx.x * blockDim.x + threadIdx.x;
  if (e >= E) return;
  int s = __builtin_nontemporal_load(src + e);
  int d = __builtin_nontemporal_load(dst + e);
  const float* f = feature + (size_t)s * 3;
  float* p = s1 + (size_t)d * 4;
  unsafeAtomicAdd(p + 0, f[0]);
  unsafeAtomicAdd(p + 1, f[1]);
  unsafeAtomicAdd(p + 2, f[2]);
  unsafeAtomicAdd(p + 3, 1.0f);
}

// ---------------------------------------------------------------------------
// Phase 3: per-node MLP + final 9->3 linear via V_WMMA_F32_16X16X4_F32.
// One wave handles a 16-node tile. z (9-wide, zero-padded to K=12) is built
// per lane; three K=4 WMMA steps accumulate D = z @ Wa + ba (bias in C).
// A-matrix layout (32-bit 16x4): lanes 0-15 hold K={0,1}, lanes 16-31 K={2,3}.
// B-matrix (4x16): N on lanes 0-15, same K split across lane halves.
// C/D: VGPR j = row j (lanes 0-15) / row j+8 (lanes 16-31), N = lane%16.
// All out-of-range B/bias terms are produced with clamped in-bounds loads +
// v_cndmask (no EXEC-masked load branches -> no divergence near WMMA).
// ---------------------------------------------------------------------------
__global__ __launch_bounds__(256) void gcn_node_kernel(
    const float4* __restrict__ s1pad, const float* __restrict__ feature,
    const float* __restrict__ x_attr, const float* __restrict__ label,
    const float* __restrict__ W0, const float* __restrict__ b0,
    const float* __restrict__ W1, const float* __restrict__ b1,
    const float* __restrict__ W2, const float* __restrict__ b2,
    const float* __restrict__ Wa, const float* __restrict__ ba,
    float* __restrict__ out, int N) {
  const int lane = threadIdx.x & 31;
  const int wave = threadIdx.x >> 5;
  const int tile = blockIdx.x * 8 + wave;  // 8 waves / block
  const int base = tile * 16;
  if (base >= N) return;  // uniform per wave: EXEC stays all-ones for WMMA

  const int m = lane & 15;   // node row within tile (and output column for B/C)
  const int hf = lane >> 4;  // lane-half selects K pairs {0,1} vs {2,3}
  const int node = base + m;
  const bool colv = (m < 3);       // this lane's output column is live
  const int nc = colv ? m : 0;     // clamped column index (always in-bounds)

  // ---- B-matrix elements: unconditional clamped loads, select to zero ----
  // Lane's K pairs per slice sl: k = 4*sl + hf*2 and k+1.
  float bw[3][2];
#pragma unroll
  for (int sl = 0; sl < 3; ++sl) {
    const int k = 4 * sl + hf * 2;
    const int k0 = k < 8 ? k : 8;          // clamp into Wa's 9 rows
    const int k1 = (k + 1) < 8 ? (k + 1) : 8;
    float wx = Wa[k0 * 3 + nc];
    float wy = Wa[k1 * 3 + nc];
    bw[sl][0] = (colv && k < 9) ? wx : 0.0f;
    bw[sl][1] = (colv && (k + 1) < 9) ? wy : 0.0f;
  }
  float bav = ba[nc];
  const float bias = colv ? bav : 0.0f;

  // ---- neighbor mean ----
  float4 s = s1pad[node];
  float inv = 1.0f / fmaxf(s.w, 1.0f);
  float h0 = s.x * inv, h1 = s.y * inv, h2 = s.z * inv;

  // ---- reduce UDF: accum = relu(label@W0 + b0 + h_agg@W1 + b1 + x@W2 + b2)
  float lab = label[node];
  float2 xa = ((const float2*)x_attr)[node];
  float f0 = feature[3 * (size_t)node + 0];
  float f1 = feature[3 * (size_t)node + 1];
  float f2 = feature[3 * (size_t)node + 2];

  float acc[3];
#pragma unroll
  for (int n = 0; n < 3; ++n) {
    float v = lab * W0[n] + h0 * W1[n] + h1 * W1[3 + n] + h2 * W1[6 + n] +
              xa.x * W2[n] + xa.y * W2[3 + n] + b0[n] + b1[n] + b2[n];
    acc[n] = fmaxf(v, 0.0f);
  }

  // z = [accum(3), h_agg(3), feature(3), pad(3)]  -- K padded 9 -> 12
  float z0 = acc[0], z1 = acc[1], z2 = acc[2];
  float z3 = h0, z4 = h1, z5 = h2;
  float z6 = f0, z7 = f1, z8 = f2;

  // ---- C accumulator preloaded with bias (column m, only m<3 meaningful)
  v8f c;
#pragma unroll
  for (int j = 0; j < 8; ++j) c[j] = bias;

  // ---- three K=4 slices of D = A(16x12) x B(12x16) + C ----
#pragma unroll
  for (int sl = 0; sl < 3; ++sl) {
    v2f A;
    if (sl == 0)      A = v2f{hf ? z2 : z0, hf ? z3 : z1};
    else if (sl == 1) A = v2f{hf ? z6 : z4, hf ? z7 : z5};
    else              A = v2f{hf ? 0.0f : z8, 0.0f};
    v2f B = v2f{bw[sl][0], bw[sl][1]};
    c = __builtin_amdgcn_wmma_f32_16x16x4_f32(
        /*neg_a=*/false, A, /*neg_b=*/false, B,
        /*c_mod=*/(short)0, c, /*reuse_a=*/false, /*reuse_b=*/false);
  }

  // ---- relu + store: lanes with column m<3 own 8 rows each ----
  if (colv) {
    const int rbase = base + hf * 8;
#pragma unroll
    for (int j = 0; j < 8; ++j)
      out[(size_t)(rbase + j) * 3 + m] = fmaxf(c[j], 0.0f);
  }
}

// ---------------------------------------------------------------------------
extern "C" void kernel_launch(void* const* d_in, const int* in_sizes, int n_in,
                              void* d_out, int out_size, void* d_ws,
                              size_t ws_size, hipStream_t stream) {
  const float* feature = (const float*)d_in[0];
  const float* x_attr  = (const float*)d_in[1];
  const float* label   = (const float*)d_in[2];
  const float* W0 = (const float*)d_in[3];
  const float* b0 = (const float*)d_in[4];
  const float* W1 = (const float*)d_in[5];
  const float* b1 = (const float*)d_in[6];
  const float* W2 = (const float*)d_in[7];
  const float* b2 = (const float*)d_in[8];
  const float* Wa = (const float*)d_in[9];
  const float* ba = (const float*)d_in[10];
  const int* src = (const int*)d_in[11];
  const int* dst = (const int*)d_in[12];

  const int N = in_sizes[2];   // label has N elements
  const int E = in_sizes[11];  // src has E elements

  // ws layout: [0,16N) s1pad float4 accumulators; [16N,32N) packed feature
  float4* s1pad = (float4*)d_ws;
  const size_t s1bytes = (size_t)N * sizeof(float4);
  const bool have_pad = ws_size >= 2 * s1bytes;

  hipMemsetAsync(d_ws, 0, s1bytes, stream);

  const int TB = 256;
  if (have_pad) {
    float4* fpad = (float4*)((char*)d_ws + s1bytes);
    gcn_pack_kernel<<<(N + TB - 1) / TB, TB, 0, stream>>>(feature, fpad, N);
    gcn_edge_kernel<<<(E + TB - 1) / TB, TB, 0, stream>>>(
        fpad, src, dst, (float*)s1pad, E);
  } else {
    gcn_edge_kernel_direct<<<(E + TB - 1) / TB, TB, 0, stream>>>(
        feature, src, dst, (float*)s1pad, E);
  }

  const int tiles = (N + 15) / 16;          // 16 nodes per wave
  const int blocks = (tiles + 7) / 8;       // 8 waves per 256-thread block
  gcn_node_kernel<<<blocks, TB, 0, stream>>>(
      s1pad, feature, x_attr, label, W0, b0, W1, b1, W2, b2, Wa, ba,
      (float*)d_out, N);
}